// VoronoiSimpleIntegrandSlang_34918084116539
// MI455X (gfx1250) — compile-verified
//
#include <hip/hip_runtime.h>

// CDNA5 / gfx1250 Voronoi nearest-site kernel.
//
// Math: d2(point n, site m) = |x_n - pos_m|^2 (per-point-shifted version of the
// reference's d2; argmin unchanged). This is a K=4 GEMM:
//   A[m][:] = [ px_m, py_m, c_m, 1 ]            (c_m = px^2+py^2)
//   B[:][n] = [ -2*x0_n, -2*x1_n, 1, |x_n|^2 ]
//   D = A*B + 0.5  => D[m][n] = |x_n - pos_m|^2 + 0.5  (strictly positive)
// computed with V_WMMA_F32_16X16X4_F32: 16 sites x 16 points per instruction.
//
// Argmin: key = (float_bits(score) & ~127) | site_id; scores positive so u32
// ordering == float ordering; per-lane v_min_u32 tree, then one shfl_xor(16)
// merges the two row-halves (rows 0-7 vs 8-15 of each tile).
//
// Site params are DMA'd into LDS with the Tensor Data Mover
// (tensor_load_to_lds + s_wait_tensorcnt) when the builtin is available.

typedef float v2f __attribute__((ext_vector_type(2)));
typedef float v8f __attribute__((ext_vector_type(8)));
typedef unsigned int u32x4 __attribute__((ext_vector_type(4)));
typedef int i32x4 __attribute__((ext_vector_type(4)));
typedef int i32x8 __attribute__((ext_vector_type(8)));

#define NSITES  100
#define SPAD    112            // padded to 7 tiles of 16
#define NTILES  7
#define NPARAMS (1 + 5 * NSITES)  // 501 floats

#if __has_builtin(__builtin_amdgcn_tensor_load_to_lds) && \
    __has_builtin(__builtin_amdgcn_s_wait_tensorcnt)
#define USE_TDM 1
#endif

__device__ __forceinline__ unsigned umin32(unsigned a, unsigned b) {
  return a < b ? a : b;
}

__global__ __launch_bounds__(256) void voronoi_wmma_kernel(
    const float* __restrict__ x,   // (P, 2)
    const float* __restrict__ p,   // (NPARAMS,)
    float* __restrict__ out,       // (P, 3)
    int P, int ngroups)
{
  __shared__ float  s_stage[NPARAMS];  // raw param block (TDM destination)
  __shared__ float  s_px[SPAD];
  __shared__ float  s_py[SPAD];
  __shared__ float  s_c [SPAD];
  __shared__ float4 s_rgb[SPAD];

  const int tid  = threadIdx.x;
  const int lane = tid & 31;
  const int wave = tid >> 5;

  // ---- stage raw params into LDS -----------------------------------------
#ifdef USE_TDM
  if (tid < 32) {  // wave 0 issues the TDM DMA (EXEC ignored by TDM)
    const unsigned long long ga = (unsigned long long)(uintptr_t)p;
    // Flat shared pointer: low 32 bits are the LDS byte offset.
    const unsigned lds_off = (unsigned)(uintptr_t)(&s_stage[0]);

    // D# group 0: count=1 (valid), lds_addr, 57-bit global_addr, type=2.
    u32x4 g0;
    g0[0] = 1u;                                            // count=1, user mode
    g0[1] = lds_off;                                       // [63:32] lds_addr
    g0[2] = (unsigned)(ga & 0xFFFFFFFFull);                // [95:64] gaddr lo
    g0[3] = ((unsigned)(ga >> 32) & 0x01FFFFFFu) | (2u << 30);  // gaddr hi | type=2

    // D# group 1: data_size=4B, tensor 501x1, tile 501x1, stride 501.
    i32x8 g1;
    g1[0] = (int)(2u << 16);            // workgroup_mask=0, data_size=2 (4B)
    g1[1] = (int)((unsigned)NPARAMS << 16);   // tensor_dim0[15:0]
    g1[2] = (int)(1u << 16);            // tensor_dim0[31:16]=0, tensor_dim1=1
    g1[3] = (int)((unsigned)NPARAMS << 16);   // tile_dim0 = 501
    g1[4] = 1;                          // tile_dim1 = 1, tile_dim2 = 0
    g1[5] = NPARAMS;                    // tensor_dim0_stride[31:0]
    g1[6] = (int)((unsigned)NPARAMS << 16);   // tensor_dim1_stride[15:0]
    g1[7] = 0;

    const i32x4 gz4 = {0, 0, 0, 0};           // groups 2/3: unused dims
    const i32x8 gz8 = {0, 0, 0, 0, 0, 0, 0, 0};
    __builtin_amdgcn_tensor_load_to_lds(g0, g1, gz4, gz4, gz8, 0);
    __builtin_amdgcn_s_wait_tensorcnt(0);
  }
#else
  if (tid < NPARAMS) s_stage[tid] = p[tid];
#endif
  __syncthreads();

  // ---- parse site constants from staged params ---------------------------
  if (tid < SPAD) {
    float px = 0.0f, py = 0.0f, c;
    float4 rgb = make_float4(0.0f, 0.0f, 0.0f, 0.0f);
    if (tid < NSITES) {
      const float* sp = s_stage + 1 + 5 * tid;
      px = sp[0];
      py = sp[1];
      rgb.x = sp[2];
      rgb.y = sp[3];
      rgb.z = sp[4];
      c = px * px + py * py;
    } else {
      c = 1.0e6f;  // dummy site: huge positive score, never wins
    }
    s_px[tid] = px;
    s_py[tid] = py;
    s_c [tid] = c;
    s_rgb[tid] = rgb;
  }
  __syncthreads();

  // ---- per-wave A tiles (site matrix, loop-invariant) --------------------
  // A (16x4 f32): lanes 0-15 hold M=lane {K0,K1}; lanes 16-31 hold {K2,K3}.
  v2f a[NTILES];
  {
    const int m = lane & 15;
#pragma unroll
    for (int t = 0; t < NTILES; ++t) {
      const int s = t * 16 + m;
      if (lane < 16) {
        a[t][0] = s_px[s];   // K=0
        a[t][1] = s_py[s];   // K=1
      } else {
        a[t][0] = s_c[s];    // K=2
        a[t][1] = 1.0f;      // K=3
      }
    }
  }

  const v8f chalf = {0.5f, 0.5f, 0.5f, 0.5f, 0.5f, 0.5f, 0.5f, 0.5f};
  const unsigned idbase = (unsigned)((lane >> 4) << 3);  // 0 or 8 (row half)

  const int wavesPerGrid = (int)(gridDim.x * (blockDim.x >> 5));
  const int g0w = (int)(blockIdx.x * (blockDim.x >> 5)) + wave;

  for (int g = g0w; g < ngroups; g += wavesPerGrid) {
    const int base = g << 4;
    const int pl   = base + (lane & 15);
    const int plc  = (pl < P) ? pl : (P - 1);

    // prefetch next group's points (global_prefetch_b8 path)
    __builtin_prefetch(x + 2 * (size_t)(plc + (wavesPerGrid << 4)), 0, 1);

    const float2 xv = reinterpret_cast<const float2*>(x)[plc];

    // B (4x16 f32): lanes 0-15 hold N=lane rows {K0,K1}; lanes 16-31 {K2,K3}.
    v2f b;
    if (lane < 16) {
      b[0] = -2.0f * xv.x;
      b[1] = -2.0f * xv.y;
    } else {
      b[0] = 1.0f;
      b[1] = xv.x * xv.x + xv.y * xv.y;  // |x|^2 of point lane-16
    }

    unsigned kmin = 0xFFFFFFFFu;
#pragma unroll
    for (int t = 0; t < NTILES; ++t) {
      // D = A*B + 0.5 : 16 sites x 16 points of (|x-pos|^2 + 0.5)
      v8f d = __builtin_amdgcn_wmma_f32_16x16x4_f32(
          false, a[t], false, b, (short)0, chalf, false, false);

      unsigned key[8];
#pragma unroll
      for (int v = 0; v < 8; ++v) {
        const unsigned bits = __float_as_uint(d[v]);
        key[v] = (bits & 0xFFFFFF80u) | (idbase | (unsigned)(t * 16 + v));
      }
      // balanced tree (ILP + v_min3 fusion)
      const unsigned m01 = umin32(key[0], key[1]);
      const unsigned m23 = umin32(key[2], key[3]);
      const unsigned m45 = umin32(key[4], key[5]);
      const unsigned m67 = umin32(key[6], key[7]);
      kmin = umin32(kmin, umin32(umin32(m01, m23), umin32(m45, m67)));
    }

    // merge row halves: lane L holds rows 0-7, lane L+16 holds rows 8-15
    const unsigned other = (unsigned)__shfl_xor((int)kmin, 16, 32);
    kmin = umin32(kmin, other);

    if ((lane < 16) && (pl < P)) {
      const int idx = (int)(kmin & 127u);
      const float4 rgb = s_rgb[idx];
      float* o = out + 3 * (size_t)pl;
      o[0] = rgb.x;
      o[1] = rgb.y;
      o[2] = rgb.z;
    }
  }
}

extern "C" void kernel_launch(void* const* d_in, const int* in_sizes, int n_in,
                              void* d_out, int out_size, void* d_ws, size_t ws_size,
                              hipStream_t stream) {
  const float* x   = (const float*)d_in[0];
  const float* p   = (const float*)d_in[1];
  float*       out = (float*)d_out;

  const int P       = in_sizes[0] / 2;     // 4,194,304
  const int ngroups = (P + 15) / 16;       // 16 points per wave-iteration
  const int wavesPerBlock = 256 / 32;

  int blocks = (ngroups + wavesPerBlock - 1) / wavesPerBlock;
  if (blocks > 16384) blocks = 16384;
  if (blocks < 1) blocks = 1;

  voronoi_wmma_kernel<<<blocks, 256, 0, stream>>>(x, p, out, P, ngroups);
}